// ProjectiveTransformerLayer_51075751084643
// MI455X (gfx1250) — compile-verified
//
#include <hip/hip_runtime.h>
#include <hip/hip_bf16.h>

// MI455X / gfx1250 projective spatial transformer.
//
// Store-bandwidth-bound kernel (201MB out @ 23.3TB/s ~ 8.6us floor). The 3MB
// source image stays resident in the 192MB L2, so bilinear gathers are cache
// traffic. The per-batch 3x3 projective matmul is computed on the matrix pipe
// with V_WMMA_F32_16X16X4_F32 (full f32, matching the reference einsum):
//   A (16x4): theta rows in M=0..2 AND duplicated in M=8..10 (K padded to 4)
//   B (4x16): 16 grid points (x_t, y_t, 1, 0) as columns
// CDNA5 D layout => lane l<16 sees rows 0..7 of col l, lane l>=16 sees rows
// 8..15 of col l-16 -> with the row duplication EVERY lane gets (T0,T1,T2) of
// its column in d[0..2] with no cross-lane traffic. Two WMMAs cover the
// wave's 32 consecutive pixels.

typedef __attribute__((ext_vector_type(2))) float v2f;
typedef __attribute__((ext_vector_type(8))) float v8f;

#define OUT_H 512
#define OUT_W 512
#define IMG_H 512
#define IMG_W 512
#define NPIX  (OUT_H * OUT_W)

__global__ __launch_bounds__(256) void proj_tx_kernel(
    const float* __restrict__ theta,  // [64, 8]
    const float* __restrict__ img,    // [512, 512, 3]
    float* __restrict__ out)          // [64, 512, 512, 3]
{
    const int lane = threadIdx.x & 31;
    const int wave = threadIdx.x >> 5;
    const int blocksPerBatch = NPIX / 256;            // 1024
    const int b    = blockIdx.x / blocksPerBatch;
    const int base = (blockIdx.x % blocksPerBatch) * 256 + wave * 32;

    const int half = lane >> 4;   // which K-pair / which row-half this lane feeds
    const int m    = lane & 15;

    // ---- A operand: 16x4 f32. Lane layout: lanes 0-15 hold K=0,1 of row m;
    //      lanes 16-31 hold K=2,3 of row m. theta row i at M=i and M=i+8.
    const float* th = theta + b * 8;
    float a0 = 0.0f, a1 = 0.0f;
    {
        int i = -1;
        if (m < 3) i = m;
        else if (m >= 8 && m < 11) i = m - 8;
        if (i >= 0) {
            const int j0 = half * 2;            // K index 0 or 2
            const int k0 = i * 3 + j0;
            a0 = (k0 < 8) ? th[k0] : 1.0f;      // theta[2][2] == 1
            if (half == 0) {                    // K=1 column (K=3 stays 0-pad)
                const int k1 = k0 + 1;
                a1 = (k1 < 8) ? th[k1] : 1.0f;
            }
        }
    }
    v2f A; A.x = a0; A.y = a1;

    // ---- B operands: 4x16 f32, columns = grid points. Lanes 0-15 hold
    //      K=0,1 (x_t, y_t) of col m; lanes 16-31 hold K=2,3 (1, 0).
    const float step = 2.0f / (float)(OUT_W - 1);    // linspace(-1,1,512)
    v2f B0, B1;
    {
        const int p0 = base + m;        // columns of WMMA #0 (pixels base..base+15)
        const int p1 = base + 16 + m;   // columns of WMMA #1 (pixels base+16..+31)
        if (half == 0) {
            B0.x = -1.0f + (float)(p0 & (OUT_W - 1)) * step;   // x_t
            B0.y = -1.0f + (float)(p0 >> 9) * step;            // y_t
            B1.x = -1.0f + (float)(p1 & (OUT_W - 1)) * step;
            B1.y = -1.0f + (float)(p1 >> 9) * step;
        } else {
            B0.x = 1.0f; B0.y = 0.0f;   // homogeneous 1, K=3 pad
            B1.x = 1.0f; B1.y = 0.0f;
        }
    }

    // ---- Matrix pipe: T = theta3x3 * [x; y; 1] for 16 columns per WMMA.
    v8f cz = {};
    v8f D0 = __builtin_amdgcn_wmma_f32_16x16x4_f32(false, A, false, B0,
                                                   (short)0, cz, false, false);
    v8f D1 = __builtin_amdgcn_wmma_f32_16x16x4_f32(false, A, false, B1,
                                                   (short)0, cz, false, false);

    // Lane l<16: rows 0..2 of col l from D0 (pixel base+l).
    // Lane l>=16: rows 8..10 of col l-16 from D1 (pixel base+l). Both land in d[0..2].
    const float T0 = (half == 0) ? D0[0] : D1[0];
    const float T1 = (half == 0) ? D0[1] : D1[1];
    const float T2 = (half == 0) ? D0[2] : D1[2];

    // ---- Perspective divide + bilinear sample (faithful to reference).
    const float z  = T2 + 1e-7f;
    float xs = T0 / z;
    float ys = T1 / z;
    xs = fminf(fmaxf(xs, -1.0f), 1.0f);
    ys = fminf(fmaxf(ys, -1.0f), 1.0f);
    const float x = (xs + 1.0f) * 0.5f * (float)(IMG_W - 1);
    const float y = (ys + 1.0f) * 0.5f * (float)(IMG_H - 1);

    const float x0f = floorf(x), y0f = floorf(y);
    const float x1f = x0f + 1.0f, y1f = y0f + 1.0f;
    const int x0 = (int)x0f;
    const int y0 = (int)y0f;
    const int x1 = (int)fminf(x1f, (float)(IMG_W - 1));
    const int y1 = (int)fminf(y1f, (float)(IMG_H - 1));

    const float w00 = (x1f - x) * (y1f - y);
    const float w01 = (x - x0f) * (y1f - y);
    const float w10 = (x1f - x) * (y - y0f);
    const float w11 = (x - x0f) * (y - y0f);

    // 3MB image: gathers hit L2/WGP$, not HBM.
    const float* r00 = img + (y0 * IMG_W + x0) * 3;
    const float* r01 = img + (y0 * IMG_W + x1) * 3;
    const float* r10 = img + (y1 * IMG_W + x0) * 3;
    const float* r11 = img + (y1 * IMG_W + x1) * 3;

    const float o0 = w00 * r00[0] + w01 * r01[0] + w10 * r10[0] + w11 * r11[0];
    const float o1 = w00 * r00[1] + w01 * r01[1] + w10 * r10[1] + w11 * r11[1];
    const float o2 = w00 * r00[2] + w01 * r01[2] + w10 * r10[2] + w11 * r11[2];

    // 32 lanes x 12B = 384 contiguous bytes per wave: coalesced stores.
    const int p = base + lane;
    float* op = out + ((size_t)b * NPIX + (size_t)p) * 3;
    op[0] = o0;
    op[1] = o1;
    op[2] = o2;
}

extern "C" void kernel_launch(void* const* d_in, const int* in_sizes, int n_in,
                              void* d_out, int out_size, void* d_ws, size_t ws_size,
                              hipStream_t stream) {
    const float* theta = (const float*)d_in[0];   // [64, 8] f32
    const float* img   = (const float*)d_in[1];   // [1, 512, 512, 3] f32
    float*       out   = (float*)d_out;           // [64, 512, 512, 3] f32

    const int B = in_sizes[0] / 8;                // 64
    const int blocksPerBatch = NPIX / 256;        // 1024
    dim3 grid(B * blocksPerBatch);                // 65536 blocks
    dim3 block(256);                              // 8 wave32s per WGP launch
    proj_tx_kernel<<<grid, block, 0, stream>>>(theta, img, out);
}